// MoELayer_50921132261643
// MI455X (gfx1250) — compile-verified
//
#include <hip/hip_runtime.h>
#include <cmath>

// Problem constants (fixed by the reference: B=2,T=2048 -> N=4096, D=1024, E=8, F=4096, top_k=2)
#define D_DIM 1024
#define E_EXP 8
#define F_DIM 4096
#define TM 16           // token tile (WMMA M)
#define FCHUNK 512      // F processed per phase (act LDS footprint)
#define NPHASE (F_DIM / FCHUNK)
#define HSTR 1028       // padded LDS row stride for h  (1028 % 64 == 4 -> conflict-free col reads)
#define ASTR 516        // padded LDS row stride for act (516 % 64 == 4)

typedef __attribute__((ext_vector_type(2))) float v2f;
typedef __attribute__((ext_vector_type(8))) float v8f;

// V_WMMA_F32_16X16X4_F32 : full-precision fp32 matrix core op (we are HBM-bound, not math-bound)
__device__ __forceinline__ v8f wmma_f32(v2f a, v2f b, v8f c) {
  // 8-arg pattern: (neg_a, A, neg_b, B, c_mod, C, reuse_a, reuse_b)
  return __builtin_amdgcn_wmma_f32_16x16x4_f32(false, a, false, b, (short)0, c, false, false);
}

__global__ void zero_cnt_kernel(int* cnt) {
  if (threadIdx.x < E_EXP) cnt[threadIdx.x] = 0;
}

// One block per token: RMS stats, router scores, top-2 softmax, per-expert gather lists,
// and residual init of the output (sum of combine weights == 1 -> out starts as x).
__global__ __launch_bounds__(256) void router_kernel(
    const float* __restrict__ x, const float* __restrict__ g_moe,
    const float* __restrict__ rw, float* __restrict__ out,
    float* __restrict__ inv_rms, int* __restrict__ cnt,
    int* __restrict__ tok_idx, float* __restrict__ tok_w, int N)
{
  int n = blockIdx.x;
  int tid = threadIdx.x;                 // 256 threads x 4 floats = D
  const float* xr = x + (size_t)n * D_DIM;
  float4 xv = ((const float4*)xr)[tid];
  ((float4*)(out + (size_t)n * D_DIM))[tid] = xv;   // residual init

  __shared__ float red[256];
  __shared__ float sc[E_EXP];

  red[tid] = xv.x*xv.x + xv.y*xv.y + xv.z*xv.z + xv.w*xv.w;
  __syncthreads();
  #pragma unroll
  for (int s = 128; s > 0; s >>= 1) {
    if (tid < s) red[tid] += red[tid + s];
    __syncthreads();
  }
  float inv = rsqrtf(red[0] / (float)D_DIM + 1e-6f);
  __syncthreads();

  float4 gv = ((const float4*)g_moe)[tid];
  float h0 = xv.x*gv.x, h1 = xv.y*gv.y, h2 = xv.z*gv.z, h3 = xv.w*gv.w;
  const float* r0 = rw + (size_t)(tid * 4) * E_EXP;
  float p[E_EXP];
  #pragma unroll
  for (int e = 0; e < E_EXP; ++e)
    p[e] = h0*r0[e] + h1*r0[E_EXP+e] + h2*r0[2*E_EXP+e] + h3*r0[3*E_EXP+e];

  for (int e = 0; e < E_EXP; ++e) {
    red[tid] = p[e];
    __syncthreads();
    #pragma unroll
    for (int s = 128; s > 0; s >>= 1) {
      if (tid < s) red[tid] += red[tid + s];
      __syncthreads();
    }
    if (tid == 0) sc[e] = red[0] * inv;
    __syncthreads();
  }

  if (tid == 0) {
    inv_rms[n] = inv;
    int b0 = 0; float v0 = sc[0];
    for (int e = 1; e < E_EXP; ++e) if (sc[e] > v0) { v0 = sc[e]; b0 = e; }
    int b1 = -1; float v1 = -3.4e38f;
    for (int e = 0; e < E_EXP; ++e) if (e != b0 && sc[e] > v1) { v1 = sc[e]; b1 = e; }
    float e1 = expf(v1 - v0);
    float s  = 1.f + e1;
    float w0 = 1.f / s;
    float w1 = e1 / s;
    int p0 = atomicAdd(&cnt[b0], 1);
    tok_idx[(size_t)b0 * N + p0] = n; tok_w[(size_t)b0 * N + p0] = w0;
    int p1 = atomicAdd(&cnt[b1], 1);
    tok_idx[(size_t)b1 * N + p1] = n; tok_w[(size_t)b1 * N + p1] = w1;
  }
}

// Fused per-expert SwiGLU FFN on a 16-token gathered tile.
// LDS: h[16][1024] (rms-normed tokens, ~66KB) + act[16][512] (~33KB) -> uses CDNA5's 320KB WGP LDS.
// 8 waves: GEMM1 wave owns a 16x64 f-slice per phase; GEMM2 wave owns a persistent 16x128 d-slice of y.
__global__ __launch_bounds__(256, 1) void expert_ffn_kernel(
    const float* __restrict__ x, const float* __restrict__ g_e,
    const float* __restrict__ w1, const float* __restrict__ w3,
    const float* __restrict__ w2, const float* __restrict__ inv_rms,
    const int* __restrict__ cnt, const int* __restrict__ tok_idx,
    const float* __restrict__ tok_w, float* __restrict__ out,
    int N, int tiles_per_e)
{
  int e    = blockIdx.x / tiles_per_e;
  int tile = blockIdx.x % tiles_per_e;
  int cnte = cnt[e];
  int m0   = tile * TM;
  if (m0 >= cnte) return;          // uniform early-exit (keeps EXEC all-ones for WMMA)

  extern __shared__ float smem[];
  float* hs = smem;                // [TM][HSTR]
  float* as = smem + TM * HSTR;    // [TM][ASTR]
  __shared__ int   toks[TM];
  __shared__ float tws[TM];
  __shared__ float tinv[TM];

  int tid = threadIdx.x;
  if (tid < TM) {
    int m = m0 + tid;
    int t; float w;
    if (m < cnte) { t = tok_idx[(size_t)e * N + m]; w = tok_w[(size_t)e * N + m]; }
    else          { t = tok_idx[(size_t)e * N];     w = 0.f; }   // pad rows: valid token, zero weight
    toks[tid] = t; tws[tid] = w; tinv[tid] = inv_rms[t];
  }
  __syncthreads();

  // Stage h = rmsnorm(x) * g_e for 16 tokens into LDS (coalesced float4 per row)
  {
    int c4 = tid * 4;
    const float* ge = g_e + (size_t)e * D_DIM + c4;
    float4 gv = *(const float4*)ge;
    #pragma unroll 1
    for (int row = 0; row < TM; ++row) {
      float4 xv = *(const float4*)(x + (size_t)toks[row] * D_DIM + c4);
      float iv = tinv[row];
      float* hp = hs + row * HSTR + c4;
      hp[0] = xv.x * iv * gv.x; hp[1] = xv.y * iv * gv.y;
      hp[2] = xv.z * iv * gv.z; hp[3] = xv.w * iv * gv.w;
    }
  }
  __syncthreads();

  int wave  = tid >> 5;
  int lane  = tid & 31;
  int l16   = lane & 15;
  int hi2   = (lane >= 16) ? 2 : 0;   // K offset for hi half of wave (f32 A/B frag layout)
  int mrow0 = (lane >= 16) ? 8 : 0;   // M offset for C frag layout

  const v8f vzero = {};
  v8f y[8];
  #pragma unroll
  for (int j = 0; j < 8; ++j) y[j] = vzero;

  const size_t wb1 = (size_t)e * D_DIM * F_DIM;
  const size_t wb2 = (size_t)e * F_DIM * D_DIM;

  for (int ph = 0; ph < NPHASE; ++ph) {
    int fbase = ph * FCHUNK;

    // ---- GEMM1: s1 = h@W1, s3 = h@W3 on this wave's 16x64 f-slice ----
    v8f s1[4], s3[4];
    #pragma unroll
    for (int j = 0; j < 4; ++j) { s1[j] = vzero; s3[j] = vzero; }
    int fw = fbase + wave * 64;
    for (int k = 0; k < D_DIM; k += 4) {
      int kk = k + hi2;
      v2f a;
      a.x = hs[l16 * HSTR + kk];
      a.y = hs[l16 * HSTR + kk + 1];
      const float* w1p = w1 + wb1 + (size_t)kk * F_DIM + fw + l16;
      const float* w3p = w3 + wb1 + (size_t)kk * F_DIM + fw + l16;
      #pragma unroll
      for (int j = 0; j < 4; ++j) {
        v2f b1; b1.x = w1p[j*16]; b1.y = w1p[j*16 + F_DIM];
        v2f b3; b3.x = w3p[j*16]; b3.y = w3p[j*16 + F_DIM];
        s1[j] = wmma_f32(a, b1, s1[j]);
        s3[j] = wmma_f32(a, b3, s3[j]);
      }
    }
    // act = silu(s1) * s3  -> LDS
    #pragma unroll
    for (int j = 0; j < 4; ++j) {
      int fcol = wave * 64 + j * 16 + l16;
      #pragma unroll
      for (int i = 0; i < 8; ++i) {
        float u = s1[j][i];
        float v = (u / (1.f + expf(-u))) * s3[j][i];
        as[(mrow0 + i) * ASTR + fcol] = v;
      }
    }
    __syncthreads();

    // ---- GEMM2 partial: y += act @ W2 over this F chunk, wave's 16x128 d-slice ----
    for (int k = 0; k < FCHUNK; k += 4) {
      int kk = k + hi2;
      v2f a;
      a.x = as[l16 * ASTR + kk];
      a.y = as[l16 * ASTR + kk + 1];
      const float* w2p = w2 + wb2 + (size_t)(fbase + kk) * D_DIM + wave * 128 + l16;
      #pragma unroll
      for (int j = 0; j < 8; ++j) {
        v2f b; b.x = w2p[j*16]; b.y = w2p[j*16 + D_DIM];
        y[j] = wmma_f32(a, b, y[j]);
      }
    }
    __syncthreads();   // protect act before next phase overwrites it
  }

  // Epilogue: out[tok][d] += combine_weight * y   (out already holds x)
  #pragma unroll
  for (int j = 0; j < 8; ++j) {
    int d = wave * 128 + j * 16 + l16;
    #pragma unroll
    for (int i = 0; i < 8; ++i) {
      int m = mrow0 + i;
      float w = tws[m];
      if (w != 0.f)
        atomicAdd(out + (size_t)toks[m] * D_DIM + d, w * y[j][i]);
    }
  }
}

extern "C" void kernel_launch(void* const* d_in, const int* in_sizes, int n_in,
                              void* d_out, int out_size, void* d_ws, size_t ws_size,
                              hipStream_t stream)
{
  const float* x     = (const float*)d_in[0];
  const float* g_moe = (const float*)d_in[1];
  const float* rw    = (const float*)d_in[2];
  const float* g_e   = (const float*)d_in[3];
  const float* w1    = (const float*)d_in[4];
  const float* w3    = (const float*)d_in[5];
  const float* w2    = (const float*)d_in[6];
  // d_in[7] = top_k (== 2, baked into router)
  float* out = (float*)d_out;

  int N = in_sizes[0] / D_DIM;   // B*T tokens

  // Workspace layout (~272KB for N=4096): counters | inv_rms[N] | tok_idx[E][N] | tok_w[E][N]
  char*  ws      = (char*)d_ws;
  int*   cnt     = (int*)ws;
  float* inv_rms = (float*)(ws + 256);
  int*   tok_idx = (int*)(ws + 256 + (size_t)N * 4);
  float* tok_w   = (float*)(ws + 256 + (size_t)N * 4 + (size_t)E_EXP * N * 4);

  zero_cnt_kernel<<<1, 32, 0, stream>>>(cnt);
  router_kernel<<<N, 256, 0, stream>>>(x, g_moe, rw, out, inv_rms, cnt, tok_idx, tok_w, N);

  int tiles = (N + TM - 1) / TM;
  size_t shmem = (size_t)(TM * HSTR + TM * ASTR) * sizeof(float);  // ~98.8 KB of WGP LDS
  expert_ffn_kernel<<<E_EXP * tiles, 256, shmem, stream>>>(
      x, g_e, w1, w3, w2, inv_rms, cnt, tok_idx, tok_w, out, N, tiles);
}